// SCTokenProcessor_53085795779497
// MI455X (gfx1250) — compile-verified
//
#include <hip/hip_runtime.h>
#include <math.h>

// ---- problem constants (from reference setup_inputs) ----
#define S_STEPS 96
#define N_TOK   1024
#define N_TILE  (N_TOK / 16)
#define NS      11          // idxs = 8,16,...,88
#define PI_F      3.14159265358979323846f
#define TWO_PI_F  6.28318530717958647693f
#define INV_2PI_F 0.15915494309189533577f
#define LN2_F     0.69314718055994530942f
#define LOG2_LN2_F (-0.52876637294489770f)   // log2(ln 2)

typedef float v2f __attribute__((ext_vector_type(2)));
typedef float v8f __attribute__((ext_vector_type(8)));

// Python-style (x+pi) mod 2pi - pi via floor (branchless)
__device__ __forceinline__ float wrapA(float x) {
    return x - TWO_PI_F * floorf((x + PI_F) * INV_2PI_F);
}

__device__ __forceinline__ unsigned pcg_hash(unsigned x) {
    x = x * 747796405u + 2891336453u;
    unsigned w = ((x >> ((x >> 28u) + 4u)) ^ x) * 277803737u;
    return (w >> 22u) ^ w;
}

// Gumbel(0,1) from 24-bit uniform, entirely in log2 space:
//   -ln(-ln u) = (-log2(ln2) - log2(-log2 u)) * ln2
// two bare v_log_f32 + 2 VALU, no OCML guard code.
__device__ __forceinline__ float gumbel24(unsigned h) {
    const float u  = (float)(h >> 8) * (1.0f / 16777216.0f) + (0.5f / 16777216.0f);
    const float l1 = __builtin_amdgcn_logf(u);     // log2(u) < 0
    const float l2 = __builtin_amdgcn_logf(-l1);   // log2(-log2 u)
    return (LOG2_LN2_F - l2) * (-LN2_F) * 1.0f == 0.0f ? 0.0f : (0.52876637f - l2) * LN2_F;
}

__global__ __launch_bounds__(256)
void SCTokenProcessor_53085795779497_kernel(const int*   __restrict__ valid,
                                            const float* __restrict__ pos,
                                            const float* __restrict__ heading,
                                            const float* __restrict__ tok_ep,
                                            float* __restrict__ out,
                                            int A)
{
    // ---- stage token table in LDS: (r^2, tx, ty, th) per token ----
    __shared__ float4 tokLDS[N_TOK];
    const int tid = threadIdx.x;
    for (int t = tid; t < N_TOK; t += blockDim.x) {
        float tx = tok_ep[t * 3 + 0];
        float ty = tok_ep[t * 3 + 1];
        float th = tok_ep[t * 3 + 2];
        tokLDS[t] = make_float4(tx * tx + ty * ty, tx, ty, th);
    }
    __syncthreads();

    const int  lane   = tid & 31;
    const int  wave   = tid >> 5;
    const int  lane16 = lane & 15;
    const bool hi     = lane >= 16;         // upper half-wave mirrors same agent
    const int  a      = blockIdx.x * 128 + wave * 16 + lane16;
    const int  base   = a * S_STEPS;

    // ---- phase 1: heading-cleaning scan (branchless, vectorized loads) ----
    unsigned vmask = 0u;                 // bit j = valid at frame 8j (post-clean/extrap)
    float hj[12];                        // cleaned heading at frames 8j
    float hp = 0.0f; int vprev = 0;
    int   fv = -1;  float hT = 0.0f;
    int   origv8 = 0;
#pragma unroll
    for (int j = 0; j < 12; ++j) {
        const int4   iv0 = *(const int4*)  (valid   + base + 8 * j);
        const int4   iv1 = *(const int4*)  (valid   + base + 8 * j + 4);
        const float4 hv0 = *(const float4*)(heading + base + 8 * j);
        const float4 hv1 = *(const float4*)(heading + base + 8 * j + 4);
        const int   vk[8] = { iv0.x != 0, iv0.y != 0, iv0.z != 0, iv0.w != 0,
                              iv1.x != 0, iv1.y != 0, iv1.z != 0, iv1.w != 0 };
        const float hk[8] = { hv0.x, hv0.y, hv0.z, hv0.w, hv1.x, hv1.y, hv1.z, hv1.w };
#pragma unroll
        for (int k = 0; k < 8; ++k) {
            const int s = j * 8 + k;
            const int vv = vk[k];
            if (s == 0) {
                hp = hk[0];
            } else {
                const float d = fabsf(wrapA(hp - hk[k]));
                const bool keep = (d > 1.5f) && vprev && vv;
                hp = keep ? hp : hk[k];
            }
            if (s == 8) origv8 = vv;
            if (k == 0) { vmask |= (unsigned)vv << j; hj[j] = hp; }
            const bool take = (fv < 0) && vv;      // first-valid frame
            fv = take ? s : fv;
            hT = take ? hp : hT;
            vprev = vv;
        }
    }

    // sample-frame positions (only the 12 frames we need) + first-valid position
    float pxj[12], pyj[12];
#pragma unroll
    for (int j = 0; j < 12; ++j) {
        const float2 p = *(const float2*)(pos + (base + 8 * j) * 2);
        pxj[j] = p.x; pyj[j] = p.y;
    }
    const float2 pT = *(const float2*)(pos + (base + (fv < 0 ? 0 : fv)) * 2);

    // ---- phase 2: stationary extrapolation override (at most one multiple of 8) ----
    {
        const int t = (fv < 0) ? 0 : fv;
        int nf = t & 7;
        if (t == 16 && !origv8) nf = 8;
        const bool doFill = (fv >= 0) && (nf > 0);
        const int  jf     = doFill ? ((t - nf) >> 3) : -1;
        if (doFill) vmask |= 1u << jf;
#pragma unroll
        for (int j = 0; j < 12; ++j) {
            const bool f = (j == jf);
            hj[j]  = f ? hT   : hj[j];
            pxj[j] = f ? pT.x : pxj[j];
            pyj[j] = f ? pT.y : pyj[j];
        }
    }

    // ---- phase 3: 11-step matching scan ----
    float gpx = pxj[0], gpy = pyj[0], gh = hj[0];   // GT carry
    float spx = pxj[0], spy = pyj[0], sh = hj[0];   // sampled carry

    const long long szA = (long long)A * NS;
    float* o_valid  = out;
    float* o_idxgt  = out + 1 * szA;
    float* o_gtpos  = out + 2 * szA;   // 2*szA elements
    float* o_gthead = out + 4 * szA;
    float* o_idxs   = out + 5 * szA;
    float* o_spos   = out + 6 * szA;   // 2*szA elements
    float* o_shead  = out + 8 * szA;

#pragma unroll
    for (int i = 0; i < NS; ++i) {
        const bool  val = ((vmask >> i) & (vmask >> (i + 1)) & 1u) != 0u;
        const float pix = pxj[i + 1], piy = pyj[i + 1], hin = hj[i + 1];

        // agent features for the rank-3 score GEMM (agent-constant term dropped:
        // argmin and Gumbel-max are shift-invariant)
        const float cg = __cosf(gh), sg = __sinf(gh);
        const float dxg = gpx - pix, dyg = gpy - piy;
        const float ug = cg * dxg + sg * dyg;
        const float vg = cg * dyg - sg * dxg;

        const float cs = __cosf(sh), ss = __sinf(sh);
        const float dxs = spx - pix, dys = spy - piy;
        const float us = cs * dxs + ss * dys;
        const float vs = cs * dys - ss * dxs;

        // B fragment (4x16, N=agent): rows [1, 2u, 2v, 0]; K0/K1 lanes 0-15, K2/K3 lanes 16-31
        v2f bg, bs;
        bg[0] = hi ? 2.0f * vg : 1.0f;  bg[1] = hi ? 0.0f : 2.0f * ug;
        bs[0] = hi ? 2.0f * vs : 1.0f;  bs[1] = hi ? 0.0f : 2.0f * us;

        float bestG = INFINITY;  int bestGi = 0;    // argmin (GT)
        float bestS = -INFINITY; int bestSi = 0;    // Gumbel-argmax (sampled)
        const unsigned seed =
            pcg_hash((unsigned)a * 1664525u ^ (unsigned)(i * 1013904223) ^ 0x9E3779B9u);
        const unsigned halfSalt = hi ? 0x68BC21EBu : 0u;

        for (int m = 0; m < N_TILE; ++m) {
            const float4 tk = tokLDS[m * 16 + lane16];
            // A fragment (16x4, M=token): row [r^2, tx, ty, 0]
            v2f af;
            af[0] = hi ? tk.z : tk.x;
            af[1] = hi ? 0.0f : tk.y;

            // Noise first (independent of WMMA -> overlaps its latency):
            // one PCG hash per tile, then a 2-VALU LCG per slot.
            unsigned ts = pcg_hash(seed ^ ((unsigned)m * 0x9E3779B9u) ^ halfSalt);
            float gum[8];
#pragma unroll
            for (int r = 0; r < 8; ++r) {
                ts = ts * 1664525u + 1013904223u;
                const float u  = (float)(ts >> 8) * (1.0f / 16777216.0f)
                               + (0.5f / 16777216.0f);          // u in (0,1) strictly
                const float l1 = __builtin_amdgcn_logf(u);      // log2(u) < 0
                const float l2 = __builtin_amdgcn_logf(-l1);    // log2(-log2 u)
                gum[r] = (0.52876637f - l2) * LN2_F;            // -ln(-ln u)
            }

            v8f cz = {};
            v8f dg  = __builtin_amdgcn_wmma_f32_16x16x4_f32(false, af, false, bg,
                                                            (short)0, cz, false, false);
            v8f dsd = __builtin_amdgcn_wmma_f32_16x16x4_f32(false, af, false, bs,
                                                            (short)0, cz, false, false);
#pragma unroll
            for (int r = 0; r < 8; ++r) {
                const int tIdx = m * 16 + r + (hi ? 8 : 0);   // D layout: M=r (+8 upper half)
                // branchless argmin; ascending order => strict '<' keeps lowest index
                const float sc = dg[r];
                const bool  cG = sc < bestG;
                bestGi = cG ? tIdx : bestGi;
                bestG  = cG ? sc   : bestG;
                // branchless Gumbel-argmax
                const float pert = gum[r] - dsd[r];
                const bool  cS = pert > bestS;
                bestSi = cS ? tIdx : bestSi;
                bestS  = cS ? pert : bestS;
            }
        }

        // combine the two half-wave token partitions (both halves end up identical)
        {
            const float oG = __shfl_xor(bestG, 16); const int oGi = __shfl_xor(bestGi, 16);
            const bool cG = (oG < bestG) || (oG == bestG && oGi < bestGi);
            bestGi = cG ? oGi : bestGi;  bestG = cG ? oG : bestG;
            const float oS = __shfl_xor(bestS, 16); const int oSi = __shfl_xor(bestSi, 16);
            const bool cS = (oS > bestS) || (oS == bestS && oSi < bestSi);
            bestSi = cS ? oSi : bestSi;  bestS = cS ? oS : bestS;
        }

        // GT carry update
        {
            const float4 tg = tokLDS[bestGi];
            const float gx  = cg * tg.y - sg * tg.z + gpx;
            const float gy  = sg * tg.y + cg * tg.z + gpy;
            const float nh  = wrapA(gh + tg.w);
            gpx = val ? gx : pix; gpy = val ? gy : piy; gh = val ? nh : hin;
        }
        // sampled carry update
        {
            const float4 tsk = tokLDS[bestSi];
            const float gxs  = cs * tsk.y - ss * tsk.z + spx;
            const float gys  = ss * tsk.y + cs * tsk.z + spy;
            const float nhs  = wrapA(sh + tsk.w);
            spx = val ? gxs : pix; spy = val ? gys : piy; sh = val ? nhs : hin;
        }

        if (!hi) {   // one writer per agent
            const long long oi = (long long)a * NS + i;
            o_valid [oi]         = val ? 1.0f : 0.0f;
            o_idxgt [oi]         = (float)bestGi;
            o_gtpos [oi * 2 + 0] = val ? gpx : 0.0f;
            o_gtpos [oi * 2 + 1] = val ? gpy : 0.0f;
            o_gthead[oi]         = val ? gh  : 0.0f;
            o_idxs  [oi]         = (float)bestSi;
            o_spos  [oi * 2 + 0] = val ? spx : 0.0f;
            o_spos  [oi * 2 + 1] = val ? spy : 0.0f;
            o_shead [oi]         = val ? sh  : 0.0f;
        }
    }
}

extern "C" void kernel_launch(void* const* d_in, const int* in_sizes, int n_in,
                              void* d_out, int out_size, void* d_ws, size_t ws_size,
                              hipStream_t stream) {
    (void)n_in; (void)out_size; (void)d_ws; (void)ws_size;
    const int*   valid   = (const int*)  d_in[0];   // (A, 96) bool->int
    const float* pos     = (const float*)d_in[1];   // (A, 96, 2) f32
    const float* heading = (const float*)d_in[2];   // (A, 96) f32
    const float* tok_ep  = (const float*)d_in[3];   // (1024, 3) f32
    float* out = (float*)d_out;

    const int A = in_sizes[0] / S_STEPS;            // 8192
    const int agentsPerBlock = 128;                 // 8 waves x 16 agents
    const int blocks = (A + agentsPerBlock - 1) / agentsPerBlock;

    SCTokenProcessor_53085795779497_kernel<<<blocks, 256, 0, stream>>>(
        valid, pos, heading, tok_ep, out, A);
}